// NetworkSubjectLayers_26852135534915
// MI455X (gfx1250) — compile-verified
//
#include <hip/hip_runtime.h>

typedef float v2f __attribute__((ext_vector_type(2)));
typedef float v8f __attribute__((ext_vector_type(8)));

#define B_   256
#define D_   2048
#define S_   8
#define OUT_ 1000

// ---------------------------------------------------------------------------
// Phase 1: bucket sample indices by subject.
// Workspace layout (ints): [0..7] = counts per subject, [8 ..] = idx[8][256].
// ---------------------------------------------------------------------------
__global__ __launch_bounds__(B_) void nsl_group_by_subject(
    const int* __restrict__ sid, int* __restrict__ ws) {
  __shared__ int cnt[S_];
  const int t = threadIdx.x;
  if (t < S_) cnt[t] = 0;
  __syncthreads();
  const int s = sid[t] & (S_ - 1);
  const int pos = atomicAdd(&cnt[s], 1);
  ws[S_ + s * B_ + pos] = t;
  __syncthreads();
  if (t < S_) ws[t] = cnt[t];
}

// ---------------------------------------------------------------------------
// Phase 2: grouped GEMM on the FP32 WMMA path (v_wmma_f32_16x16x4_f32).
// One wave per block. Tile: 32 rows (2 M-fragments) x 64 cols (4 N-fragments).
// grid.x = ceil(OUT/64) = 16, grid.y = S_ * 8 (8 subjects x 8 M-supertiles).
// ---------------------------------------------------------------------------
__global__ __launch_bounds__(32) void nsl_grouped_gemm_wmma(
    const float* __restrict__ x, const float* __restrict__ W,
    const float* __restrict__ bias, const int* __restrict__ ws,
    float* __restrict__ out) {
  const int s     = blockIdx.y >> 3;
  const int mt    = blockIdx.y & 7;
  const int cnt   = ws[s];
  const int rbase = mt * 32;
  if (rbase >= cnt) return;                 // uniform per wave: EXEC stays all-1s
  const int* __restrict__ idx = ws + S_ + s * B_;

  const int lane = threadIdx.x;
  const int nrow = lane & 15;               // A: M row; B/C/D: N column
  const int kh   = lane >> 4;               // selects K pair {2kh, 2kh+1} / M half

  // Gather x row pointers for the two 16-row A fragments (pad with a valid row).
  const int r0 = rbase + nrow;
  const int r1 = rbase + 16 + nrow;
  const int smp0 = idx[(r0 < cnt) ? r0 : rbase];
  const int smp1 = idx[(r1 < cnt) ? r1 : rbase];
  const float* __restrict__ xr0 = x + (size_t)smp0 * D_;
  const float* __restrict__ xr1 = x + (size_t)smp1 * D_;

  // Per-lane output columns for the 4 N-fragments (clamped for loads).
  const int ncol = blockIdx.x * 64 + nrow;
  int ccol[4];
#pragma unroll
  for (int t = 0; t < 4; ++t) {
    const int c = ncol + 16 * t;
    ccol[t] = (c < OUT_) ? c : (OUT_ - 1);
  }

  const float* __restrict__ Wp = W + (size_t)s * D_ * OUT_;

  v8f acc[2][4];
#pragma unroll
  for (int m = 0; m < 2; ++m)
#pragma unroll
    for (int t = 0; t < 4; ++t) acc[m][t] = (v8f){0.f,0.f,0.f,0.f,0.f,0.f,0.f,0.f};

  for (int k = 0; k < D_; k += 4) {
    const int kk = k + 2 * kh;
    // A fragments: contiguous float2 per lane.
    const v2f a0 = { xr0[kk], xr0[kk + 1] };
    const v2f a1 = { xr1[kk], xr1[kk + 1] };
    const float* __restrict__ wk = Wp + (size_t)kk * OUT_;
#pragma unroll
    for (int t = 0; t < 4; ++t) {
      // B fragment: rows kk, kk+1 at this lane's column (lanes 0-15 contiguous).
      const v2f bf = { wk[ccol[t]], wk[OUT_ + ccol[t]] };
      acc[0][t] = __builtin_amdgcn_wmma_f32_16x16x4_f32(
          false, a0, false, bf, (short)0, acc[0][t], false, false);
      acc[1][t] = __builtin_amdgcn_wmma_f32_16x16x4_f32(
          false, a1, false, bf, (short)0, acc[1][t], false, false);
    }
  }

  // Store: C/D layout -> lane holds (M = i + 8*kh, N = lane%16) in VGPR i.
#pragma unroll
  for (int mh = 0; mh < 2; ++mh) {
#pragma unroll
    for (int i = 0; i < 8; ++i) {
      const int m = i + 8 * kh;
      const int r = rbase + mh * 16 + m;
      if (r >= cnt) continue;
      const int smp = idx[r];
#pragma unroll
      for (int t = 0; t < 4; ++t) {
        const int c = ncol + 16 * t;
        if (c < OUT_)
          out[(size_t)smp * OUT_ + c] = acc[mh][t][i] + bias[s * OUT_ + c];
      }
    }
  }
}

// ---------------------------------------------------------------------------
extern "C" void kernel_launch(void* const* d_in, const int* in_sizes, int n_in,
                              void* d_out, int out_size, void* d_ws, size_t ws_size,
                              hipStream_t stream) {
  const float* x    = (const float*)d_in[0];   // (B, D) fp32
  const int*   sid  = (const int*)d_in[1];     // (B,) integer
  const float* W    = (const float*)d_in[2];   // (S, D, OUT) fp32
  const float* bias = (const float*)d_in[3];   // (S, OUT) fp32
  float* out = (float*)d_out;                  // (B, OUT) fp32
  int*   ws  = (int*)d_ws;                     // counts[8] + idx[8][256]

  nsl_group_by_subject<<<1, B_, 0, stream>>>(sid, ws);

  dim3 grid((OUT_ + 63) / 64, S_ * 8);         // 16 x 64 single-wave blocks
  nsl_grouped_gemm_wmma<<<grid, 32, 0, stream>>>(x, W, bias, ws, out);
}